// ResonanceLayer_27066883900192
// MI455X (gfx1250) — compile-verified
//
#include <hip/hip_runtime.h>
#include <hip/hip_bf16.h>
#include <math.h>

// ---------------------------------------------------------------------------
// ResonanceLayer for MI455X (gfx1250, wave32, WMMA).
// y = LN(2*x + sparse_irfft(delta)),  delta at 256 bins f_k = floor(k*512/255)
// All dense transforms done with V_WMMA_F32_16X16X4_F32 (full fp32 precision;
// problem is HBM-bound at 23.3 TB/s, so fp32 matrix rate is not the limiter).
// ---------------------------------------------------------------------------

#define BATCH 8
#define SEQ   2048
#define DIM   1024
#define KF    256
#define NPAD  2048          // padded_len == SEQ (power of two)
#define LN_EPS 1e-5f

typedef __attribute__((ext_vector_type(2))) float v2f;
typedef __attribute__((ext_vector_type(8))) float v8f;

// D = A(16x4) * B(4x16) + C(16x16), fp32, wave32.
__device__ __forceinline__ v8f wmma_f32(v2f a, v2f b, v8f c) {
  // (neg_a, A, neg_b, B, c_mod, C, reuse_a, reuse_b); f32 has no A/B neg.
  return __builtin_amdgcn_wmma_f32_16x16x4_f32(false, a, false, b,
                                               (short)0, c, false, false);
}

// ---------------------------------------------------------------------------
// K1: trig tables. cosKT[k][t] = cos(2*pi*f_k*t/2048), nsinKT = -sin(...).
// f_k*t reduced mod 2048 exactly in integers -> no large-angle error.
// ---------------------------------------------------------------------------
__global__ void trig_kernel(float* __restrict__ cosKT, float* __restrict__ nsinKT) {
  int idx = blockIdx.x * blockDim.x + threadIdx.x;   // KF*SEQ total
  int k = idx >> 11;           // /2048
  int t = idx & 2047;
  int f = (k * 512) / 255;     // matches np.linspace(0,512,256).astype(int64)
  int ph = (f * t) & (NPAD - 1);
  float theta = (float)ph * (6.283185307179586f / (float)NPAD);
  float s, c;
  sincosf(theta, &s, &c);
  cosKT[idx]  = c;
  nsinKT[idx] = -s;
}

// ---------------------------------------------------------------------------
// K2: forward sparse DFT.  Xre[b,k,d] = sum_t cos[k][t]*x[b,t,d]
//                          Xim[b,k,d] = sum_t (-sin[k][t])*x[b,t,d]
// grid = (DIM/128, KF/16, BATCH), block = 256 (8 waves, 16 cols each).
// ---------------------------------------------------------------------------
__global__ void fwd_kernel(const float* __restrict__ x,
                           const float* __restrict__ cosKT,
                           const float* __restrict__ nsinKT,
                           float* __restrict__ Xre, float* __restrict__ Xim) {
  const int tid  = threadIdx.x;
  const int wave = tid >> 5;
  const int lane = tid & 31;
  const int l15  = lane & 15;
  const int h    = lane >> 4;          // half-wave select
  const int kb2  = 2 * h;              // K-pair offset within WMMA step

  const int n0   = blockIdx.x * 128 + wave * 16 + l15;   // output column (d)
  const int mrow = blockIdx.y * 16 + l15;                // frequency row (k)
  const float* xb = x + (size_t)blockIdx.z * SEQ * DIM;

  v8f accR = {0.f,0.f,0.f,0.f,0.f,0.f,0.f,0.f};
  v8f accI = {0.f,0.f,0.f,0.f,0.f,0.f,0.f,0.f};

#pragma unroll 8
  for (int t = 0; t < SEQ; t += 4) {
    const int ka = t + kb2;
    v2f ac = *(const v2f*)(cosKT  + (size_t)mrow * SEQ + ka);   // 8B contiguous
    v2f as = *(const v2f*)(nsinKT + (size_t)mrow * SEQ + ka);
    v2f bx;
    bx.x = xb[(size_t)ka       * DIM + n0];
    bx.y = xb[(size_t)(ka + 1) * DIM + n0];
    accR = wmma_f32(ac, bx, accR);
    accI = wmma_f32(as, bx, accI);
  }

  float* pre = Xre + (size_t)blockIdx.z * KF * DIM;
  float* pim = Xim + (size_t)blockIdx.z * KF * DIM;
#pragma unroll
  for (int v = 0; v < 8; ++v) {
    const int row = blockIdx.y * 16 + v + 8 * h;
    pre[(size_t)row * DIM + n0] = accR[v];
    pim[(size_t)row * DIM + n0] = accI[v];
  }
}

// ---------------------------------------------------------------------------
// K3a: complex weight + imag-delta.
//   cw = mag*(cos(ph) + i sin(ph))
//   Re_w = Xre*cwr - Xim*cwi            (kept, feeds interference GEMM)
//   di   = s_k*(Xre*cwi + Xim*cwr - Xim)   [di(0)=0: irfft ignores imag(bin0)]
// di overwrites Xim in place.
// ---------------------------------------------------------------------------
__global__ void weight_kernel(const float* __restrict__ Xre,
                              float* __restrict__ XimDi,
                              const float* __restrict__ mag,
                              const float* __restrict__ ph,
                              float* __restrict__ Rew) {
  const size_t idx = (size_t)blockIdx.x * blockDim.x + threadIdx.x; // B*KF*DIM
  const int r = (int)(idx % (KF * DIM));
  const int k = r >> 10;               // /DIM
  float xre = Xre[idx];
  float xim = XimDi[idx];
  float s, c;
  sincosf(ph[r], &s, &c);
  const float m   = mag[r];
  const float cwr = m * c, cwi = m * s;
  const float rew = xre * cwr - xim * cwi;
  const float imw = xre * cwi + xim * cwr;
  const float sk  = (k == 0 ? 1.0f : 2.0f) * (1.0f / (float)NPAD);
  Rew[idx]   = rew;
  XimDi[idx] = (k == 0) ? 0.0f : sk * (imw - xim);
}

// ---------------------------------------------------------------------------
// K3b: interference GEMM + real-delta epilogue.
//   I[b,j,d] = sum_k W[k,j] * Re_w[b,k,d]
//   dr[j,d]  = s_j*(Re_w[j,d] + 0.1*I[j,d] - Xre[j,d])   (overwrites Xre)
// grid = (DIM/128, KF/16, BATCH), block = 256.
// ---------------------------------------------------------------------------
__global__ void intf_kernel(const float* __restrict__ W,
                            const float* __restrict__ Rew,
                            float* __restrict__ XreDr) {
  const int tid  = threadIdx.x;
  const int wave = tid >> 5;
  const int lane = tid & 31;
  const int l15  = lane & 15;
  const int h    = lane >> 4;
  const int kb2  = 2 * h;

  const int n0   = blockIdx.x * 128 + wave * 16 + l15;
  const int jrow = blockIdx.y * 16 + l15;
  const float* rewb = Rew + (size_t)blockIdx.z * KF * DIM;

  v8f acc = {0.f,0.f,0.f,0.f,0.f,0.f,0.f,0.f};
#pragma unroll 8
  for (int k = 0; k < KF; k += 4) {
    const int ka = k + kb2;
    v2f a;                                   // A[j][k] = W[k][j] (transposed)
    a.x = W[(size_t)ka       * KF + jrow];
    a.y = W[(size_t)(ka + 1) * KF + jrow];
    v2f b;
    b.x = rewb[(size_t)ka       * DIM + n0];
    b.y = rewb[(size_t)(ka + 1) * DIM + n0];
    acc = wmma_f32(a, b, acc);
  }

#pragma unroll
  for (int v = 0; v < 8; ++v) {
    const int j = blockIdx.y * 16 + v + 8 * h;
    const size_t o = (size_t)blockIdx.z * KF * DIM + (size_t)j * DIM + n0;
    const float sk = (j == 0 ? 1.0f : 2.0f) * (1.0f / (float)NPAD);
    XreDr[o] = sk * (Rew[o] + 0.1f * acc[v] - XreDr[o]);
  }
}

// ---------------------------------------------------------------------------
// K4: inverse sparse irfft + residual.
//   y[b,t,d] = 2*x[b,t,d] + sum_k ( dr[k,d]*cos[k][t] + di[k,d]*(-sin[k][t]) )
// grid = (DIM/128, SEQ/16, BATCH), block = 256.
// ---------------------------------------------------------------------------
__global__ void inv_kernel(const float* __restrict__ x,
                           const float* __restrict__ cosKT,
                           const float* __restrict__ nsinKT,
                           const float* __restrict__ Dr,
                           const float* __restrict__ Di,
                           float* __restrict__ out) {
  const int tid  = threadIdx.x;
  const int wave = tid >> 5;
  const int lane = tid & 31;
  const int l15  = lane & 15;
  const int h    = lane >> 4;
  const int kb2  = 2 * h;

  const int n0   = blockIdx.x * 128 + wave * 16 + l15;
  const int trow = blockIdx.y * 16 + l15;
  const float* drb = Dr + (size_t)blockIdx.z * KF * DIM;
  const float* dib = Di + (size_t)blockIdx.z * KF * DIM;

  v8f acc = {0.f,0.f,0.f,0.f,0.f,0.f,0.f,0.f};
#pragma unroll 8
  for (int k = 0; k < KF; k += 4) {
    const int ka = k + kb2;
    v2f ac, an;                               // transposed trig reads
    ac.x = cosKT [(size_t)ka       * SEQ + trow];
    ac.y = cosKT [(size_t)(ka + 1) * SEQ + trow];
    an.x = nsinKT[(size_t)ka       * SEQ + trow];
    an.y = nsinKT[(size_t)(ka + 1) * SEQ + trow];
    v2f bdr, bdi;
    bdr.x = drb[(size_t)ka       * DIM + n0];
    bdr.y = drb[(size_t)(ka + 1) * DIM + n0];
    bdi.x = dib[(size_t)ka       * DIM + n0];
    bdi.y = dib[(size_t)(ka + 1) * DIM + n0];
    acc = wmma_f32(ac, bdr, acc);
    acc = wmma_f32(an, bdi, acc);
  }

#pragma unroll
  for (int v = 0; v < 8; ++v) {
    const int t = blockIdx.y * 16 + v + 8 * h;
    const size_t o = (size_t)blockIdx.z * SEQ * DIM + (size_t)t * DIM + n0;
    out[o] = acc[v] + 2.0f * x[o];
  }
}

// ---------------------------------------------------------------------------
// K5: in-place LayerNorm over last dim (1024). One block per (b,t) row.
// block = 256 threads * float4 = 1024 elements.
// ---------------------------------------------------------------------------
__global__ void ln_kernel(float* __restrict__ y,
                          const float* __restrict__ gamma,
                          const float* __restrict__ beta) {
  const int tid = threadIdx.x;
  const size_t base = (size_t)blockIdx.x * DIM;
  float4 v = ((const float4*)(y + base))[tid];

  float s  = v.x + v.y + v.z + v.w;
  float s2 = v.x*v.x + v.y*v.y + v.z*v.z + v.w*v.w;
#pragma unroll
  for (int off = 16; off > 0; off >>= 1) {     // wave32 butterfly
    s  += __shfl_xor(s,  off, 32);
    s2 += __shfl_xor(s2, off, 32);
  }
  __shared__ float sh[8], sh2[8];
  const int wv = tid >> 5, ln = tid & 31;
  if (ln == 0) { sh[wv] = s; sh2[wv] = s2; }
  __syncthreads();
  if (tid == 0) {
    float a = 0.f, b = 0.f;
#pragma unroll
    for (int i = 0; i < 8; ++i) { a += sh[i]; b += sh2[i]; }
    sh[0] = a; sh2[0] = b;
  }
  __syncthreads();
  const float mean = sh[0] * (1.0f / (float)DIM);
  const float var  = sh2[0] * (1.0f / (float)DIM) - mean * mean;
  const float rstd = rsqrtf(var + LN_EPS);

  const float4 g  = ((const float4*)gamma)[tid];
  const float4 bt = ((const float4*)beta)[tid];
  v.x = (v.x - mean) * rstd * g.x + bt.x;
  v.y = (v.y - mean) * rstd * g.y + bt.y;
  v.z = (v.z - mean) * rstd * g.z + bt.z;
  v.w = (v.w - mean) * rstd * g.w + bt.w;
  ((float4*)(y + base))[tid] = v;
}

// ---------------------------------------------------------------------------
extern "C" void kernel_launch(void* const* d_in, const int* in_sizes, int n_in,
                              void* d_out, int out_size, void* d_ws, size_t ws_size,
                              hipStream_t stream) {
  const float* x    = (const float*)d_in[0];  // (B,S,D)
  const float* mag  = (const float*)d_in[1];  // (K,D)
  const float* ph   = (const float*)d_in[2];  // (K,D)
  const float* W    = (const float*)d_in[3];  // (K,K)
  const float* gmm  = (const float*)d_in[4];  // (D,)
  const float* bta  = (const float*)d_in[5];  // (D,)
  float* out = (float*)d_out;

  // workspace layout (floats): cos | -sin | Xre(->dr) | Xim(->di) | Re_w
  float* wsf    = (float*)d_ws;
  float* cosKT  = wsf;                               // KF*SEQ
  float* nsinKT = cosKT  + (size_t)KF * SEQ;         // KF*SEQ
  float* Xre    = nsinKT + (size_t)KF * SEQ;         // B*KF*DIM
  float* Xim    = Xre    + (size_t)BATCH * KF * DIM; // B*KF*DIM
  float* Rew    = Xim    + (size_t)BATCH * KF * DIM; // B*KF*DIM

  // K1: trig tables (KF*SEQ = 524288 elems)
  trig_kernel<<<dim3((KF * SEQ) / 256), dim3(256), 0, stream>>>(cosKT, nsinKT);

  // K2: forward transform
  fwd_kernel<<<dim3(DIM / 128, KF / 16, BATCH), dim3(256), 0, stream>>>(
      x, cosKT, nsinKT, Xre, Xim);

  // K3a: complex weighting + imag delta (in-place on Xim)
  weight_kernel<<<dim3((BATCH * KF * DIM) / 256), dim3(256), 0, stream>>>(
      Xre, Xim, mag, ph, Rew);

  // K3b: interference GEMM + real delta (in-place on Xre)
  intf_kernel<<<dim3(DIM / 128, KF / 16, BATCH), dim3(256), 0, stream>>>(
      W, Rew, Xre);

  // K4: inverse transform + residual (2x)
  inv_kernel<<<dim3(DIM / 128, SEQ / 16, BATCH), dim3(256), 0, stream>>>(
      x, cosKT, nsinKT, Xre, Xim, out);

  // K5: LayerNorm in place on out
  ln_kernel<<<dim3(BATCH * SEQ), dim3(256), 0, stream>>>(out, gmm, bta);

  (void)in_sizes; (void)n_in; (void)out_size; (void)ws_size;
}